// SimpleGCN_5308579578320
// MI455X (gfx1250) — compile-verified
//
#include <hip/hip_runtime.h>
#include <hip/hip_bf16.h>
#include <math.h>

typedef float v2f __attribute__((ext_vector_type(2)));
typedef float v8f __attribute__((ext_vector_type(8)));

#define NEG_SLOPE 0.2f

// ---------------------------------------------------------------------------
// FP32 WMMA GEMM: C[M,Ncols] = A[M,K] @ B[K,Ncols]
// Each wave computes a 16 x (16*CT) super-tile with CT accumulators so the
// A operand is fetched ONCE per K-step (one global_load_b64) and feeds CT
// v_wmma_f32_16x16x4_f32 ops. block = 128 (4 waves, 4 M-tiles);
// grid = (ceil(M/64), Ncols/(16*CT)).
// ---------------------------------------------------------------------------
template <int CT>
__global__ void gemm_wmma_f32(const float* __restrict__ A,
                              const float* __restrict__ B,
                              float* __restrict__ C,
                              int M, int K, int Ncols) {
    const int wave  = threadIdx.x >> 5;
    const int lane  = threadIdx.x & 31;
    const int mtile = blockIdx.x * 4 + wave;
    const int row0  = mtile * 16;
    if (row0 >= M) return;                 // wave-uniform exit: EXEC stays all-1s

    const int col0  = blockIdx.y * (16 * CT);
    const int lrow  = lane & 15;
    const int khalf = (lane >> 4) * 2;     // lanes 0-15 hold K={0,1}, 16-31 hold K={2,3}

    int arow = row0 + lrow;
    if (arow >= M) arow = M - 1;           // clamp (no divergence); store guarded later
    const float* __restrict__ arp = A + (size_t)arow * K;

    v8f acc[CT];
#pragma unroll
    for (int t = 0; t < CT; ++t) acc[t] = (v8f){0.f,0.f,0.f,0.f,0.f,0.f,0.f,0.f};

    for (int k = 0; k < K; k += 4) {
        // one 8-byte A fetch per lane per K-step (8B aligned: K%4==0, khalf even)
        const v2f a = *(const v2f*)(arp + k + khalf);

        v2f b[CT];
#pragma unroll
        for (int t = 0; t < CT; ++t) {
            const int c = col0 + 16 * t + lrow;
            b[t].x = B[(size_t)(k + khalf)     * Ncols + c];
            b[t].y = B[(size_t)(k + khalf + 1) * Ncols + c];
        }
#pragma unroll
        for (int t = 0; t < CT; ++t) {
            // 8 args: (neg_a, A, neg_b, B, c_mod, C, reuse_a, reuse_b)
            acc[t] = __builtin_amdgcn_wmma_f32_16x16x4_f32(
                false, a, false, b[t], (short)0, acc[t], false, false);
        }
    }

    const int rbase = row0 + (lane >> 4) * 8;
#pragma unroll
    for (int t = 0; t < CT; ++t) {
        const int ccol = col0 + 16 * t + lrow;
#pragma unroll
        for (int r = 0; r < 8; ++r) {
            const int row = rbase + r;
            if (row < M) C[(size_t)row * Ncols + ccol] = acc[t][r];
        }
    }
}

static inline void launch_gemm(const float* A, const float* B, float* C,
                               int M, int K, int Ncols, hipStream_t stream) {
    if ((Ncols & 63) == 0) {
        dim3 g((M + 63) / 64, Ncols / 64);
        gemm_wmma_f32<4><<<g, 128, 0, stream>>>(A, B, C, M, K, Ncols);
    } else {
        dim3 g((M + 63) / 64, Ncols / 32);
        gemm_wmma_f32<2><<<g, 128, 0, stream>>>(A, B, C, M, K, Ncols);
    }
}

// ---------------------------------------------------------------------------
// Per-node attention logits + per-layer state init.
// ---------------------------------------------------------------------------
__global__ void attn_init(const float* __restrict__ h,
                          const float* __restrict__ a_src,
                          const float* __restrict__ a_dst,
                          float* __restrict__ als, float* __restrict__ ald,
                          float* __restrict__ emax, float* __restrict__ denom,
                          float* __restrict__ outbuf, int Nn, int dout) {
    const int i = blockIdx.x * blockDim.x + threadIdx.x;
    if (i >= Nn) return;
    const float* hr = h + (size_t)i * dout;
    float s = 0.f, d = 0.f;
    for (int f = 0; f < dout; ++f) {
        const float v = hr[f];
        s += v * a_src[f];
        d += v * a_dst[f];
    }
    als[i] = s;
    ald[i] = d;
    emax[i]  = -INFINITY;
    denom[i] = 0.f;
    float* orow = outbuf + (size_t)i * dout;
    for (int f = 0; f < dout; ++f) orow[f] = 0.f;
}

__device__ inline void atomicMaxF(float* addr, float val) {
    unsigned int* ua = (unsigned int*)addr;
    unsigned int cur = *ua;
    while (__uint_as_float(cur) < val) {
        const unsigned int assumed = cur;
        cur = atomicCAS(ua, assumed, __float_as_uint(val));
        if (cur == assumed) break;
    }
}

// edge id >= E encodes the self loop (src = dst = id - E)
__global__ void edge_max(const int* __restrict__ ei, int E, int Etot,
                         const float* __restrict__ als, const float* __restrict__ ald,
                         float* __restrict__ ebuf, float* __restrict__ emax) {
    const int e = blockIdx.x * blockDim.x + threadIdx.x;
    if (e >= Etot) return;
    int s, d;
    if (e < E) { s = ei[e]; d = ei[E + e]; } else { s = d = e - E; }
    float v = als[s] + ald[d];
    v = (v > 0.f) ? v : NEG_SLOPE * v;     // leaky_relu
    ebuf[e] = v;
    atomicMaxF(&emax[d], v);
}

__global__ void edge_exp(const int* __restrict__ ei, int E, int Etot,
                         const float* __restrict__ emax,
                         float* __restrict__ ebuf, float* __restrict__ denom) {
    const int e = blockIdx.x * blockDim.x + threadIdx.x;
    if (e >= Etot) return;
    const int d = (e < E) ? ei[E + e] : (e - E);
    const float ex = __expf(ebuf[e] - emax[d]);
    ebuf[e] = ex;
    atomicAdd(&denom[d], ex);
}

// wave-per-edge scatter of alpha * h[src] into out[dst]; lanes stride features.
__global__ void edge_scatter(const int* __restrict__ ei, int E, int Etot,
                             const float* __restrict__ ebuf,
                             const float* __restrict__ denom,
                             const float* __restrict__ h,
                             float* __restrict__ outbuf, int dout) {
    const int lane = threadIdx.x & 31;
    const int widx = blockIdx.x * (blockDim.x >> 5) + (threadIdx.x >> 5);
    if (widx >= Etot) return;
    int s, d;
    if (widx < E) { s = ei[widx]; d = ei[E + widx]; } else { s = d = widx - E; }
    const float alpha = ebuf[widx] / denom[d];
    const float* hr   = h + (size_t)s * dout;
    float*       orow = outbuf + (size_t)d * dout;
    for (int f = lane; f < dout; f += 32)
        atomicAdd(&orow[f], alpha * hr[f]);   // global_atomic_add_f32
}

__global__ void bias_elu(const float* __restrict__ in, const float* __restrict__ b,
                         float* __restrict__ out, int total, int dout) {
    const int i = blockIdx.x * blockDim.x + threadIdx.x;
    if (i >= total) return;
    const float v = in[i] + b[i % dout];
    out[i] = (v > 0.f) ? v : (__expf(v) - 1.f);  // ELU
}

// ---------------------------------------------------------------------------
// Global mean pool over 64 graphs (feature dim 256).
// ---------------------------------------------------------------------------
__global__ void pool_init(float* __restrict__ psum, float* __restrict__ pcnt) {
    const int i = blockIdx.x * blockDim.x + threadIdx.x;
    if (i < 64 * 256) psum[i] = 0.f;
    if (i < 64) pcnt[i] = 0.f;
}

__global__ void pool_acc(const float* __restrict__ x, const int* __restrict__ batch,
                         float* __restrict__ psum, float* __restrict__ pcnt, int Nn) {
    const int lane = threadIdx.x & 31;
    const int node = blockIdx.x * (blockDim.x >> 5) + (threadIdx.x >> 5);
    if (node >= Nn) return;
    const int g = batch[node];
    const float* xr = x + (size_t)node * 256;
    for (int f = lane; f < 256; f += 32)
        atomicAdd(&psum[g * 256 + f], xr[f]);
    if (lane == 0) atomicAdd(&pcnt[g], 1.f);
}

__global__ void pool_div(float* __restrict__ psum, const float* __restrict__ pcnt) {
    const int i = blockIdx.x * blockDim.x + threadIdx.x;
    if (i >= 64 * 256) return;
    psum[i] /= fmaxf(pcnt[i >> 8], 1.f);
}

// Final tiny GEMM: out[64,10] = g[64,128] @ W[128,10] + b
__global__ void final_lin(const float* __restrict__ g, const float* __restrict__ W,
                          const float* __restrict__ b, float* __restrict__ out) {
    const int i = blockIdx.x * blockDim.x + threadIdx.x;
    if (i >= 64 * 10) return;
    const int r = i / 10, c = i % 10;
    float acc = b[c];
    for (int k = 0; k < 128; ++k)
        acc += g[r * 128 + k] * W[k * 10 + c];
    out[i] = acc;
}

// ---------------------------------------------------------------------------
extern "C" void kernel_launch(void* const* d_in, const int* in_sizes, int n_in,
                              void* d_out, int out_size, void* d_ws, size_t ws_size,
                              hipStream_t stream) {
    const int dims[5] = {128, 32, 64, 128, 256};

    const float* x0    = (const float*)d_in[0];
    const int*   ei    = (const int*)d_in[1];
    const int*   batch = (const int*)d_in[2];

    const float *Wl[4], *asrc[4], *adst[4], *bl[4];
    for (int l = 0; l < 4; ++l) {
        Wl[l]   = (const float*)d_in[3 + 4 * l];
        asrc[l] = (const float*)d_in[4 + 4 * l];
        adst[l] = (const float*)d_in[5 + 4 * l];
        bl[l]   = (const float*)d_in[6 + 4 * l];
    }
    const float* lin1_w = (const float*)d_in[19];
    const float* lin1_b = (const float*)d_in[20];
    const float* lin2_w = (const float*)d_in[21];
    const float* lin2_b = (const float*)d_in[22];

    const int N    = in_sizes[0] / 128;
    const int E    = in_sizes[1] / 2;
    const int Etot = E + N;               // implicit self loops
    const size_t NN = (size_t)N;

    // workspace layout (floats)
    float* ws    = (float*)d_ws;
    float* X     = ws;                    // N*256 : layer input/output features
    float* H     = X + NN * 256;          // N*256 : h = x @ W
    float* O     = H + NN * 256;          // N*256 : aggregated messages
    float* als   = O + NN * 256;          // N
    float* ald   = als + NN;              // N
    float* emax  = ald + NN;              // N
    float* denom = emax + NN;             // N
    float* ebuf  = denom + NN;            // Etot
    float* psum  = ebuf + Etot;           // 64*256 (becomes pooled g)
    float* pcnt  = psum + 64 * 256;       // 64
    float* mid   = pcnt + 64;             // 64*128
    float* mid2  = mid + 64 * 128;        // 64*128

    const float* xin = x0;
    for (int l = 0; l < 4; ++l) {
        const int din = dims[l], dout = dims[l + 1];

        launch_gemm(xin, Wl[l], H, N, din, dout, stream);

        attn_init<<<(N + 255) / 256, 256, 0, stream>>>(
            H, asrc[l], adst[l], als, ald, emax, denom, O, N, dout);

        edge_max<<<(Etot + 255) / 256, 256, 0, stream>>>(ei, E, Etot, als, ald, ebuf, emax);
        edge_exp<<<(Etot + 255) / 256, 256, 0, stream>>>(ei, E, Etot, emax, ebuf, denom);
        edge_scatter<<<(Etot + 7) / 8, 256, 0, stream>>>(ei, E, Etot, ebuf, denom, H, O, dout);

        bias_elu<<<((N * dout) + 255) / 256, 256, 0, stream>>>(O, bl[l], X, N * dout, dout);
        xin = X;
    }

    // global mean pool -> psum[64,256]
    pool_init<<<(64 * 256 + 255) / 256, 256, 0, stream>>>(psum, pcnt);
    pool_acc<<<(N + 7) / 8, 256, 0, stream>>>(X, batch, psum, pcnt, N);
    pool_div<<<(64 * 256 + 255) / 256, 256, 0, stream>>>(psum, pcnt);

    // lin1: [64,256] @ [256,128] -> mid, then bias+ELU -> mid2
    launch_gemm(psum, lin1_w, mid, 64, 256, 128, stream);
    bias_elu<<<(64 * 128 + 255) / 256, 256, 0, stream>>>(mid, lin1_b, mid2, 64 * 128, 128);

    // lin2: [64,128] @ [128,10] + b -> out
    final_lin<<<(64 * 10 + 255) / 256, 256, 0, stream>>>(mid2, lin2_w, lin2_b, (float*)d_out);
}